// SimpleRAN_81578608820450
// MI455X (gfx1250) — compile-verified
//
#include <hip/hip_runtime.h>

// ---------------------------------------------------------------------------
// RAN (recurrent additive network), 2 layers, T=512 B=64 H=512, fp32.
// Fused recurrence:  g = [c_t | x_t] @ [w_hh ; w_ih] + b   (K = 1024)
// V_WMMA_F32_16X16X4_F32, 32 persistent workgroups, one grid barrier/step.
// Weight slice (128.5 KB) parked in LDS for the whole kernel (CDNA5 320KB
// LDS); [c|x_t] chunks double-buffered in LDS and filled with CDNA5 async
// global->LDS copies (ASYNCcnt) overlapped with WMMA compute.
// ---------------------------------------------------------------------------

typedef __attribute__((ext_vector_type(2))) float v2f;
typedef __attribute__((ext_vector_type(4))) float f4;
typedef __attribute__((ext_vector_type(8))) float v8f;
typedef __attribute__((ext_vector_type(4))) int   v4i;

typedef __attribute__((address_space(1))) v4i g_v4i;   // global int4
typedef __attribute__((address_space(3))) v4i l_v4i;   // LDS int4

#define T_STEPS 512
#define BATCH   64
#define HID     512
#define NGATE   1024          // 2*H
#define NBLK    32            // persistent workgroups (each owns 16 h-columns)
#define BLKT    256           // 8 waves of 32
#define KCHUNK  32
#define NKB     32            // (H_c + H_x) / KCHUNK = 1024/32
#define APAD    36            // lds_a row stride (floats): bank skew
#define BSTRIDE 1028          // lds_b per-column stride (dwords)
#define HSTRIDE 514           // lds_b per-lane-half stride (dwords)

#if __has_builtin(__builtin_amdgcn_global_load_async_to_lds_b128)
#define HAVE_ASYNC_LDS 1
#else
#define HAVE_ASYNC_LDS 0
#endif

__device__ __forceinline__ void async_copy16(void* dst_lds, const void* src_glob) {
#if HAVE_ASYNC_LDS
  __builtin_amdgcn_global_load_async_to_lds_b128(
      (g_v4i*)src_glob, (l_v4i*)dst_lds, /*offset=*/0, /*cpol=*/0);
#else
  *(f4*)dst_lds = *(const f4*)src_glob;   // sync fallback
#endif
}

__device__ __forceinline__ void async_wait0() {
#if HAVE_ASYNC_LDS
#if __has_builtin(__builtin_amdgcn_s_wait_asynccnt)
  __builtin_amdgcn_s_wait_asynccnt(0);
#else
  asm volatile("s_wait_asynccnt 0x0" ::: "memory");
#endif
#endif
}

__device__ __forceinline__ float sigmoidf_(float v) {
  return 1.0f / (1.0f + __expf(-v));
}
__device__ __forceinline__ float tanhf_(float v) {
  float e2 = __expf(2.0f * v);
  return (e2 - 1.0f) / (e2 + 1.0f);
}

__device__ __forceinline__ void grid_barrier(unsigned* cnt, unsigned target) {
  __threadfence();
  __syncthreads();
  if (threadIdx.x == 0) {
    __hip_atomic_fetch_add(cnt, 1u, __ATOMIC_ACQ_REL, __HIP_MEMORY_SCOPE_AGENT);
    unsigned v;
    do {
      __builtin_amdgcn_s_sleep(1);
      v = __hip_atomic_load(cnt, __ATOMIC_ACQUIRE, __HIP_MEMORY_SCOPE_AGENT);
    } while (v < target);
  }
  __syncthreads();
}

__global__ __launch_bounds__(BLKT, 1)
void ran_scan_layer(const float* __restrict__ x,      // [T,B,H] layer input
                    float* __restrict__ out,          // [T,B,H] layer output
                    const float* __restrict__ whh,    // [H, 2H]
                    const float* __restrict__ wih,    // [H, 2H]
                    const float* __restrict__ bias,   // [2H]
                    float* __restrict__ cb0,          // [B,H] c ping
                    float* __restrict__ cb1,          // [B,H] c pong
                    const int* __restrict__ length,   // [B]
                    float* __restrict__ cT,           // [B,H] final state out
                    unsigned* __restrict__ bar) {
  // 154.5 KB static LDS (CDNA5 WGP: 320 KB)
  __shared__ float lds_b[32 * BSTRIDE];        // 128.5 KB weight slice
  __shared__ float lds_a[2][BATCH][APAD];      //  18  KB double-buffered chunk
  __shared__ float lds_f[BATCH * 16];          //   4  KB f-gate exchange
  __shared__ float lds_i[BATCH * 16];          //   4  KB i-gate exchange

  const int tid   = threadIdx.x;
  const int wave  = tid >> 5;
  const int lane  = tid & 31;
  const int hi    = lane >> 4;          // 0: lanes 0-15, 1: lanes 16-31
  const int l15   = lane & 15;
  const int j     = blockIdx.x;         // owns h columns [16j, 16j+16)
  const int hbase = j * 16;
  const int m0    = (wave & 3) * 16;    // batch-row strip of this wave's tile
  const int isI   = wave >> 2;          // waves 0-3: f gate, 4-7: i gate
  const int colIdx = isI * 16 + l15;    // 0..31 within this WG's column slice
  const int ncol  = (isI ? HID : 0) + hbase + l15;
  const float breg = bias[ncol];

  // ---- one-time: park this WG's weight slice [1024 K x 32 cols] in LDS ----
  // per-(col,hi) contiguous pair streams: element k of column c lives at
  // dword  c*1028 + ((k>>1)&1)*514 + (k>>2)*2 + (k&1)
  for (int i = 0; i < 128; ++i) {
    int e    = tid + i * BLKT;          // 0..32767
    int kg   = e >> 5;                  // 0..1023
    int cidx = e & 31;
    int n    = ((cidx >> 4) ? HID : 0) + hbase + (cidx & 15);
    const float* wsel = (kg < HID) ? whh : wih;
    int krow = kg & (HID - 1);
    int dst  = cidx * BSTRIDE + ((kg >> 1) & 1) * HSTRIDE + (kg >> 2) * 2 + (kg & 1);
    lds_b[dst] = wsel[(size_t)krow * NGATE + n];
  }
  __syncthreads();

  // per-thread t-invariant state for the elementwise phase
  int  eidx[4], lb[4];
#pragma unroll
  for (int q = 0; q < 4; ++q) {
    eidx[q] = tid + q * BLKT;           // 0..1023 over [64 x 16]
    lb[q]   = length[eidx[q] >> 4];
  }

  for (int t = 0; t < T_STEPS; ++t) {
    const float* cbr = (t & 1) ? cb1 : cb0;
    float*       cbw = (t & 1) ? cb0 : cb1;
    const float* xt  = x + (size_t)t * (BATCH * HID);

    // launder the (t-invariant) LDS weight offset so LICM cannot hoist the
    // 256 ds_loads out of the t-loop and spill them to scratch
    unsigned boff = (unsigned)(colIdx * BSTRIDE + hi * HSTRIDE);
    asm volatile("" : "+v"(boff));
    const v2f* bp = (const v2f*)(&lds_b[boff]);   // contiguous K-pair stream

    v8f acc = {};  // one 16x16 f32 tile per wave

    // ---- software pipeline: async-copy chunk kb+1 while computing kb -----
    {
#pragma unroll
      for (int i = 0; i < 2; ++i) {     // issue chunk 0 (from c)
        int u = tid + i * BLKT;         // 0..511 float4 slots (64x32/4)
        int row = u >> 3, c4 = (u & 7) << 2;
        async_copy16(&lds_a[0][row][c4], cbr + row * HID + c4);
      }
    }

    for (int kb = 0; kb < NKB; ++kb) {
      async_wait0();                    // my share of chunk kb landed
      __syncthreads();                  // everyone's share landed; buffer free

      if (kb + 1 < NKB) {               // issue chunk kb+1 (overlaps compute)
        int kn = kb + 1;
        const float* nsrc = (kn < 16) ? (cbr + kn * KCHUNK)
                                      : (xt + (kn - 16) * KCHUNK);
        bool pf = (kn >= 16) && (t + 1 < T_STEPS);
#pragma unroll
        for (int i = 0; i < 2; ++i) {
          int u = tid + i * BLKT;
          int row = u >> 3, c4 = (u & 7) << 2;
          async_copy16(&lds_a[kn & 1][row][c4], nsrc + row * HID + c4);
          if (pf)  // prefetch next timestep's x chunk
            __builtin_prefetch(nsrc + BATCH * HID + row * HID + c4, 0, 1);
        }
      }

      // --- WMMA on chunk kb (pure LDS operands, batched 4-deep) -----------
      const float* arow = &lds_a[kb & 1][m0 + l15][2 * hi];
      const int qb = kb * (KCHUNK / 4);
#pragma unroll
      for (int kk = 0; kk < KCHUNK; kk += 16) {
        int q = qb + (kk >> 2);
        v2f a0 = *(const v2f*)(arow + kk);
        v2f a1 = *(const v2f*)(arow + kk + 4);
        v2f a2 = *(const v2f*)(arow + kk + 8);
        v2f a3 = *(const v2f*)(arow + kk + 12);
        v2f b0 = bp[q + 0], b1 = bp[q + 1], b2 = bp[q + 2], b3 = bp[q + 3];
        acc = __builtin_amdgcn_wmma_f32_16x16x4_f32(false, a0, false, b0,
                                                    (short)0, acc, false, false);
        acc = __builtin_amdgcn_wmma_f32_16x16x4_f32(false, a1, false, b1,
                                                    (short)0, acc, false, false);
        acc = __builtin_amdgcn_wmma_f32_16x16x4_f32(false, a2, false, b2,
                                                    (short)0, acc, false, false);
        acc = __builtin_amdgcn_wmma_f32_16x16x4_f32(false, a3, false, b3,
                                                    (short)0, acc, false, false);
      }
    }

    // --- bias add + stash gate tiles into LDS ------------------------------
    {
      float* gdst = isI ? lds_i : lds_f;
#pragma unroll
      for (int r = 0; r < 8; ++r) {
        int row = m0 + r + hi * 8;      // C/D layout: vgpr r -> rows r, r+8
        gdst[row * 16 + l15] = acc[r] + breg;
      }
    }
    __syncthreads();

    // --- elementwise RAN cell update on owned columns ----------------------
#pragma unroll
    for (int q = 0; q < 4; ++q) {
      int e  = eidx[q];
      int b  = e >> 4;
      int h  = hbase + (e & 15);
      float fg = lds_f[e];
      float ig = lds_i[e];
      float co = cbr[b * HID + h];
      float xv = xt[b * HID + h];
      float c1 = sigmoidf_(ig) * xv + sigmoidf_(fg) * tanhf_(co);
      float cm = (t < lb[q]) ? c1 : co;   // length mask blend
      cbw[b * HID + h] = cm;
      out[(size_t)t * (BATCH * HID) + b * HID + h] = cm;
      if (t == T_STEPS - 1) cT[b * HID + h] = cm;
    }

    grid_barrier(bar, (unsigned)(NBLK * (t + 1)));
  }
}

// ---------------------------------------------------------------------------
extern "C" void kernel_launch(void* const* d_in, const int* in_sizes, int n_in,
                              void* d_out, int out_size, void* d_ws,
                              size_t ws_size, hipStream_t stream) {
  (void)in_sizes; (void)n_in; (void)out_size; (void)ws_size;

  const float* input = (const float*)d_in[0];   // [T,B,H]
  const float* hx    = (const float*)d_in[1];   // [L,B,H]
  const int*   len   = (const int*)d_in[2];     // [B]
  const float* wih0  = (const float*)d_in[3];
  const float* whh0  = (const float*)d_in[4];
  const float* b0    = (const float*)d_in[5];
  const float* wih1  = (const float*)d_in[6];
  const float* whh1  = (const float*)d_in[7];
  const float* b1    = (const float*)d_in[8];

  float* out = (float*)d_out;
  float* xout = out;                                   // [T,B,H]
  float* cT0  = out + (size_t)T_STEPS * BATCH * HID;   // c_n[0]
  float* cT1  = cT0 + BATCH * HID;                     // c_n[1]

  char* ws = (char*)d_ws;
  float* x1  = (float*)ws;                             // layer-0 output, 64 MB
  float* cb0 = (float*)(ws + (size_t)T_STEPS * BATCH * HID * sizeof(float));
  float* cb1 = cb0 + BATCH * HID;
  unsigned* bar = (unsigned*)(cb1 + BATCH * HID);

  const size_t cbytes = (size_t)BATCH * HID * sizeof(float);

  // ---- layer 0: input -> x1 ----
  (void)hipMemcpyAsync(cb0, hx, cbytes, hipMemcpyDeviceToDevice, stream);
  (void)hipMemsetAsync(bar, 0, sizeof(unsigned), stream);
  ran_scan_layer<<<NBLK, BLKT, 0, stream>>>(input, x1, whh0, wih0, b0,
                                            cb0, cb1, len, cT0, bar);

  // ---- layer 1: x1 -> d_out ----
  (void)hipMemcpyAsync(cb0, hx + (size_t)BATCH * HID, cbytes,
                       hipMemcpyDeviceToDevice, stream);
  (void)hipMemsetAsync(bar, 0, sizeof(unsigned), stream);
  ran_scan_layer<<<NBLK, BLKT, 0, stream>>>(x1, xout, whh1, wih1, b1,
                                            cb0, cb1, len, cT1, bar);
}